// Uni_62088047231179
// MI455X (gfx1250) — compile-verified
//
#include <hip/hip_runtime.h>
#include <stdint.h>

#define N_NODES 200000
#define N_EDGES 6400000
#define T_STEPS 10
#define N_BLOCKSB 12
#define CHUNK 2048                       // edges staged per workgroup via TDM
#define EDGE_GRID (N_EDGES / CHUNK)      // 3125, exact

typedef float v2f __attribute__((ext_vector_type(2)));
typedef float v8f __attribute__((ext_vector_type(8)));
typedef unsigned int v4u __attribute__((ext_vector_type(4)));
typedef int v4i __attribute__((ext_vector_type(4)));
typedef int v8i __attribute__((ext_vector_type(8)));

// ---------------------------------------------------------------------------
// TDM: 1-D tile load (elems * 4 bytes) global -> LDS, per CDNA5 D# layout
// (CDNA5 ISA). clang-23 6-arg builtin form:
// (uint32x4 g0, int32x8 g1, int32x4 g2, int32x4 g3, int32x8 extra, i32 cpol)
// ---------------------------------------------------------------------------
__device__ __forceinline__ void tdm_load_1d(const void* gptr, uint32_t lds_off,
                                            int elems) {
  uint64_t ga = (uint64_t)(uintptr_t)gptr;
  v4u g0;
  g0.x = 1u;                                   // count=1, user descriptor
  g0.y = lds_off;                              // lds_addr (bytes)
  g0.z = (uint32_t)ga;                         // global_addr[31:0]
  g0.w = (uint32_t)(ga >> 32) | (2u << 30);    // global_addr[56:32] | type=2
  v8i g1;
  uint32_t e = (uint32_t)elems;
  g1[0] = (int)(2u << 16);                     // data_size = 4 bytes
  g1[1] = (int)((e & 0xFFFFu) << 16);          // tensor_dim0[15:0] -> bits[63:48]
  g1[2] = (int)((e >> 16) | (1u << 16));       // tensor_dim0[31:16] | tensor_dim1=1
  g1[3] = (int)((e & 0xFFFFu) << 16);          // tile_dim0 -> bits[127:112]
  g1[4] = 1;                                   // tile_dim1=1, tile_dim2=0
  g1[5] = (int)e;                              // tensor_dim0_stride[31:0]
  g1[6] = 0;                                   // stride hi | tensor_dim1_stride lo
  g1[7] = 0;
  v4i zz4 = {0, 0, 0, 0};
  v8i zz8 = {0, 0, 0, 0, 0, 0, 0, 0};
  __builtin_amdgcn_tensor_load_to_lds(g0, g1, zz4, zz4, zz8, 0);
}

// ---------------------------------------------------------------------------
// Setup kernels
// ---------------------------------------------------------------------------
__global__ __launch_bounds__(256) void deg_init_k(float* __restrict__ deg) {
  int i = blockIdx.x * 256 + threadIdx.x;
  if (i < N_NODES) deg[i] = 1.0f;              // self-loop
}

__global__ __launch_bounds__(256) void deg_accum_k(const int* __restrict__ dst,
                                                   float* __restrict__ deg) {
  int i = blockIdx.x * 256 + threadIdx.x;      // 4 edges per thread
  int4 d = ((const int4*)dst)[i];
  __hip_atomic_fetch_add(&deg[d.x], 1.0f, __ATOMIC_RELAXED, __HIP_MEMORY_SCOPE_AGENT);
  __hip_atomic_fetch_add(&deg[d.y], 1.0f, __ATOMIC_RELAXED, __HIP_MEMORY_SCOPE_AGENT);
  __hip_atomic_fetch_add(&deg[d.z], 1.0f, __ATOMIC_RELAXED, __HIP_MEMORY_SCOPE_AGENT);
  __hip_atomic_fetch_add(&deg[d.w], 1.0f, __ATOMIC_RELAXED, __HIP_MEMORY_SCOPE_AGENT);
}

__global__ __launch_bounds__(256) void state_init_k(
    const float* __restrict__ x, const float* __restrict__ deg,
    float* __restrict__ dinv, float2* __restrict__ t, float2* __restrict__ z,
    float2* __restrict__ p) {
  int i = blockIdx.x * 256 + threadIdx.x;
  if (i >= N_NODES) return;
  float dv = rsqrtf(deg[i]);                   // deg >= 1 always
  dinv[i] = dv;
  float2 v; v.x = x[i * 8 + 3]; v.y = 0.0f;    // xr = x[:,3], xi = 0
  t[i] = v; z[i] = v;
  float2 zero; zero.x = 0.0f; zero.y = 0.0f;
  p[i] = zero;
}

__global__ __launch_bounds__(256) void norm_k(const int* __restrict__ src,
                                              const int* __restrict__ dst,
                                              const float* __restrict__ dinv,
                                              float* __restrict__ w) {
  int i = blockIdx.x * 256 + threadIdx.x;      // 4 edges per thread
  int4 s = ((const int4*)src)[i];
  int4 d = ((const int4*)dst)[i];
  float4 o;
  o.x = dinv[s.x] * dinv[d.x];
  o.y = dinv[s.y] * dinv[d.y];
  o.z = dinv[s.z] * dinv[d.z];
  o.w = dinv[s.w] * dinv[d.w];
  ((float4*)w)[i] = o;
}

// ---------------------------------------------------------------------------
// Fused complex SpMV scatter: p[dst] += w * t[src]  (real+imag together).
// Edge chunk (src,dst,w) staged into LDS by the Tensor Data Mover.
// ---------------------------------------------------------------------------
__global__ __launch_bounds__(256) void spmv_edge_k(
    const int* __restrict__ src, const int* __restrict__ dst,
    const float* __restrict__ w, const float2* __restrict__ t,
    float2* __restrict__ p) {
  __shared__ int s_src[CHUNK];
  __shared__ int s_dst[CHUNK];
  __shared__ float s_w[CHUNK];
  const int tid = threadIdx.x;
  const long long base = (long long)blockIdx.x * CHUNK;
  if (tid < 32) {                              // wave 0 drives the DMA
    tdm_load_1d(src + base, (uint32_t)(uintptr_t)&s_src[0], CHUNK);
    tdm_load_1d(dst + base, (uint32_t)(uintptr_t)&s_dst[0], CHUNK);
    tdm_load_1d(w + base, (uint32_t)(uintptr_t)&s_w[0], CHUNK);
    __builtin_amdgcn_s_wait_tensorcnt(0);
  }
  __syncthreads();
#pragma unroll
  for (int j = 0; j < CHUNK / 256; ++j) {
    int e = j * 256 + tid;                     // bank-conflict-free LDS reads
    int si = s_src[e];
    int di = s_dst[e];
    float wv = s_w[e];
    float2 tv = t[si];                         // one b64 gather (L2 resident)
    float* pd = (float*)&p[di];
    __hip_atomic_fetch_add(pd, wv * tv.x, __ATOMIC_RELAXED, __HIP_MEMORY_SCOPE_AGENT);
    __hip_atomic_fetch_add(pd + 1, wv * tv.y, __ATOMIC_RELAXED, __HIP_MEMORY_SCOPE_AGENT);
  }
}

// ---------------------------------------------------------------------------
// Per-node step: add diagonal (self-loop dinv^2), multiply by i/t, accumulate
// into z, reset p for the next step.
// ---------------------------------------------------------------------------
__global__ __launch_bounds__(256) void node_step_k(
    const float* __restrict__ dinv, float2* __restrict__ t,
    float2* __restrict__ z, float2* __restrict__ p, float inv_t) {
  int i = blockIdx.x * 256 + threadIdx.x;
  if (i >= N_NODES) return;
  float dv = dinv[i];
  float d2 = dv * dv;
  float2 tv = t[i];
  float2 pv = p[i];
  float pr = pv.x + d2 * tv.x;
  float pi = pv.y + d2 * tv.y;
  float ntr = -pi * inv_t;
  float nti = pr * inv_t;
  float2 nt; nt.x = ntr; nt.y = nti;
  t[i] = nt;
  float2 zv = z[i];
  zv.x += ntr; zv.y += nti;
  z[i] = zv;
  float2 zero; zero.x = 0.0f; zero.y = 0.0f;
  p[i] = zero;
}

// ---------------------------------------------------------------------------
// Block rotation via V_WMMA_F32_16X16X4_F32: x = R(theta_b) * z, then
// t = z = x for the next block. 16 nodes per wave.
// A (16x4, 2 VGPR): lanes 0-15 hold (zr,zi) in K0,K1; lanes 16-31 (K2,K3)=0.
// B (4x16, 2 VGPR): v0 = row K0 {c,s,0..}, v1 = row K1 {-s,c,0..}.
// D (16x16, 8 VGPR): col n lives in lane n (rows 0-7) / lane 16+n (rows 8-15),
// so xr is in lanes 0/16, xi in lanes 1/17.
// ---------------------------------------------------------------------------
__global__ __launch_bounds__(256) void rotate_wmma_k(
    const float* __restrict__ theta, int b, float2* __restrict__ t,
    float2* __restrict__ z) {
  const int lane = threadIdx.x & 31;
  const int wv = threadIdx.x >> 5;
  const int base = (blockIdx.x * 8 + wv) * 16;
  const int li = lane & 15;
  int node = base + li;
  int cn = node < N_NODES ? node : (N_NODES - 1);  // clamp: keep EXEC full
  float2 zv = z[cn];
  bool lo = lane < 16;
  v2f A;
  A.x = lo ? zv.x : 0.0f;
  A.y = lo ? zv.y : 0.0f;
  float th = theta[b];
  float c = cosf(th), s = sinf(th);
  v2f B;
  B.x = (lane == 0) ? c : ((lane == 1) ? s : 0.0f);
  B.y = (lane == 0) ? -s : ((lane == 1) ? c : 0.0f);
  v8f C = {0.0f, 0.0f, 0.0f, 0.0f, 0.0f, 0.0f, 0.0f, 0.0f};
  v8f D = __builtin_amdgcn_wmma_f32_16x16x4_f32(false, A, false, B, (short)0, C,
                                                false, false);
  int rb = base + ((lane & 16) ? 8 : 0);
  if (lane == 0 || lane == 16) {               // real column (N=0)
#pragma unroll
    for (int j = 0; j < 8; ++j) {
      int n = rb + j;
      if (n < N_NODES) { float v = D[j]; t[n].x = v; z[n].x = v; }
    }
  } else if (lane == 1 || lane == 17) {        // imag column (N=1)
#pragma unroll
    for (int j = 0; j < 8; ++j) {
      int n = rb + j;
      if (n < N_NODES) { float v = D[j]; t[n].y = v; z[n].y = v; }
    }
  }
}

__global__ __launch_bounds__(256) void extract_k(const float2* __restrict__ t,
                                                 float* __restrict__ out) {
  int i = blockIdx.x * 256 + threadIdx.x;
  if (i < N_NODES) out[i] = t[i].x;
}

// ---------------------------------------------------------------------------
extern "C" void kernel_launch(void* const* d_in, const int* in_sizes, int n_in,
                              void* d_out, int out_size, void* d_ws,
                              size_t ws_size, hipStream_t stream) {
  const float* x = (const float*)d_in[0];
  const int* ei = (const int*)d_in[1];
  const float* theta = (const float*)d_in[2];
  const int* src = ei;
  const int* dst = ei + N_EDGES;

  char* ws = (char*)d_ws;
  float* deg = (float*)ws;          ws += (size_t)N_NODES * sizeof(float);
  float* dinv = (float*)ws;         ws += (size_t)N_NODES * sizeof(float);
  float* w = (float*)ws;            ws += (size_t)N_EDGES * sizeof(float);
  float2* t = (float2*)ws;          ws += (size_t)N_NODES * sizeof(float2);
  float2* z = (float2*)ws;          ws += (size_t)N_NODES * sizeof(float2);
  float2* p = (float2*)ws;          ws += (size_t)N_NODES * sizeof(float2);

  const int nbN = (N_NODES + 255) / 256;       // 782
  const int nbE4 = N_EDGES / 4 / 256;          // 6250, exact
  const int nbRot = (N_NODES + 127) / 128;     // 1563 (8 waves x 16 nodes)

  deg_init_k<<<nbN, 256, 0, stream>>>(deg);
  deg_accum_k<<<nbE4, 256, 0, stream>>>(dst, deg);
  state_init_k<<<nbN, 256, 0, stream>>>(x, deg, dinv, t, z, p);
  norm_k<<<nbE4, 256, 0, stream>>>(src, dst, dinv, w);

  for (int b = 0; b < N_BLOCKSB; ++b) {
    for (int step = 1; step <= T_STEPS; ++step) {
      spmv_edge_k<<<EDGE_GRID, 256, 0, stream>>>(src, dst, w, t, p);
      node_step_k<<<nbN, 256, 0, stream>>>(dinv, t, z, p, 1.0f / (float)step);
    }
    rotate_wmma_k<<<nbRot, 256, 0, stream>>>(theta, b, t, z);
  }
  extract_k<<<nbN, 256, 0, stream>>>(t, (float*)d_out);
}